// Sample_GCN_4801773437668
// MI455X (gfx1250) — compile-verified
//
#include <hip/hip_runtime.h>

typedef float v2f __attribute__((ext_vector_type(2)));
typedef float v8f __attribute__((ext_vector_type(8)));

#define IN_DIM 256
#define HID    128
#define OUTD   16

// ---------- degree / norm ----------
__global__ void k_init_deg(float* deg, int n) {
  int i = blockIdx.x * blockDim.x + threadIdx.x;
  if (i < n) deg[i] = 1.0f;  // self-loop weight
}

__global__ void k_deg_accum(const int* dst, const float* w, float* deg, int E) {
  int e = blockIdx.x * blockDim.x + threadIdx.x;
  if (e < E) atomicAdd(&deg[dst[e]], w[e]);
}

__global__ void k_dinv(float* deg, int n) {
  int i = blockIdx.x * blockDim.x + threadIdx.x;
  if (i < n) {
    float d = deg[i];
    deg[i] = (d > 0.0f) ? rsqrtf(d) : 0.0f;
  }
}

__global__ void k_norm(const int* src, const int* dst, const float* w,
                       const float* dinv, float* norm, int E) {
  int e = blockIdx.x * blockDim.x + threadIdx.x;
  if (e < E) norm[e] = dinv[src[e]] * w[e] * dinv[dst[e]];
}

// ---------- GEMM1: h1[N,128] = x[N,256] @ W1[256,128], fp32 WMMA ----------
// 256 threads = 8 waves; block owns 16 rows; wave w owns cols [16w,16w+16)
#define XS_LD (IN_DIM + 4)  // stride%64==4 -> rows 0..15 hit distinct banks; 16B aligned
__global__ void k_gemm1(const float* __restrict__ x, const float* __restrict__ W1,
                        float* __restrict__ h1, int N) {
  __shared__ float xs[16 * XS_LD];
  const int tid  = threadIdx.x;
  const int base = blockIdx.x * 16;

  // stage 16x256 x-tile via float4 (coalesced b128)
  for (int i = tid; i < 16 * (IN_DIM / 4); i += 256) {
    int r = i >> 6, c4 = i & 63;
    int g = base + r;
    float4 v = make_float4(0.f, 0.f, 0.f, 0.f);
    if (g < N) v = ((const float4*)(x + (size_t)g * IN_DIM))[c4];
    *(float4*)&xs[r * XS_LD + c4 * 4] = v;
  }
  __syncthreads();

  const int wave = tid >> 5, lane = tid & 31;
  const int half = lane >> 4, m = lane & 15;  // m = A-row / B-col / D-col
  const int col0 = wave * 16;

  v8f c = {};
  for (int k = 0; k < IN_DIM; k += 4) {
    int kk = k + 2 * half;
    v2f a, b;
    a.x = xs[m * XS_LD + kk];
    a.y = xs[m * XS_LD + kk + 1];
    b.x = W1[(size_t)kk * HID + col0 + m];
    b.y = W1[(size_t)(kk + 1) * HID + col0 + m];
    c = __builtin_amdgcn_wmma_f32_16x16x4_f32(false, a, false, b, (short)0, c,
                                              false, false);
  }

  if (base + 16 <= N) {  // full tile: unguarded stores (always taken, N%16==0)
    float* hp = h1 + (size_t)(base + 8 * half) * HID + col0 + m;
#pragma unroll
    for (int v = 0; v < 8; ++v) hp[(size_t)v * HID] = c[v];
  } else {
#pragma unroll
    for (int v = 0; v < 8; ++v) {
      int row = base + v + 8 * half;
      if (row < N) h1[(size_t)row * HID + col0 + m] = c[v];
    }
  }
}

// ---------- out1 = b1 + dinv^2 * h1 (self loop + bias), float4 ----------
__global__ void k_init_out1(const float4* h1, const float* dinv, const float4* b1,
                            float4* out1, int total4) {
  int i = blockIdx.x * blockDim.x + threadIdx.x;
  if (i < total4) {
    int node = i >> 5, c4 = i & 31;  // HID/4 = 32
    float di = dinv[node];
    float s = di * di;
    float4 h = h1[i], b = b1[c4];
    out1[i] = make_float4(b.x + s * h.x, b.y + s * h.y, b.z + s * h.z, b.w + s * h.w);
  }
}

// ---------- edge scatter layer 1: one wave per edge, float4 gather ----------
__global__ void k_scatter1(const int* __restrict__ src, const int* __restrict__ dst,
                           const float* __restrict__ norm,
                           const float* __restrict__ h1, float* __restrict__ out1,
                           int E) {
  int e = blockIdx.x * 8 + (threadIdx.x >> 5);
  if (e >= E) return;  // wave-uniform exit
  int lane = threadIdx.x & 31;
  int s = src[e], d = dst[e];
  float nm = norm[e];
  float4 v = ((const float4*)(h1 + (size_t)s * HID))[lane];  // b128, 512B/wave
  float* op = out1 + (size_t)d * HID + lane * 4;
  atomicAdd(op + 0, nm * v.x);
  atomicAdd(op + 1, nm * v.y);
  atomicAdd(op + 2, nm * v.z);
  atomicAdd(op + 3, nm * v.w);
}

// ---------- GEMM2: h3[N,16] = relu(out1)[N,128] @ W2[128,16], fp32 WMMA ----------
__global__ void k_gemm2(const float* __restrict__ out1, const float* __restrict__ W2,
                        float* __restrict__ h3, int N) {
  __shared__ float ws2[HID * OUTD];  // 8 KB
  const int tid = threadIdx.x;
  for (int i = tid; i < HID * OUTD; i += 256) ws2[i] = W2[i];
  __syncthreads();

  const int wave = tid >> 5, lane = tid & 31;
  const int half = lane >> 4, m = lane & 15;
  const int base = (blockIdx.x * 8 + wave) * 16;

  v8f c = {};
  const int arow = base + m;
  const bool aok = (arow < N);
  const float* ap = out1 + (size_t)(aok ? arow : 0) * HID;
  for (int k = 0; k < HID; k += 4) {
    int kk = k + 2 * half;
    v2f a, b;
    a.x = aok ? fmaxf(ap[kk], 0.0f) : 0.0f;  // fused ReLU
    a.y = aok ? fmaxf(ap[kk + 1], 0.0f) : 0.0f;
    b.x = ws2[kk * OUTD + m];
    b.y = ws2[(kk + 1) * OUTD + m];
    c = __builtin_amdgcn_wmma_f32_16x16x4_f32(false, a, false, b, (short)0, c,
                                              false, false);
  }

  if (base + 16 <= N) {  // full tile fast path
    float* hp = h3 + (size_t)(base + 8 * half) * OUTD + m;
#pragma unroll
    for (int v = 0; v < 8; ++v) hp[(size_t)v * OUTD] = c[v];
  } else {
#pragma unroll
    for (int v = 0; v < 8; ++v) {
      int row = base + v + 8 * half;
      if (row < N) h3[(size_t)row * OUTD + m] = c[v];
    }
  }
}

// ---------- out = b2 + dinv^2 * h3, float4 ----------
__global__ void k_init_out2(const float4* h3, const float* dinv, const float4* b2,
                            float4* out, int total4) {
  int i = blockIdx.x * blockDim.x + threadIdx.x;
  if (i < total4) {
    int node = i >> 2, c4 = i & 3;  // OUTD/4 = 4
    float di = dinv[node];
    float s = di * di;
    float4 h = h3[i], b = b2[c4];
    out[i] = make_float4(b.x + s * h.x, b.y + s * h.y, b.z + s * h.z, b.w + s * h.w);
  }
}

// ---------- edge scatter layer 2: 4 threads per edge, float4 gather ----------
__global__ void k_scatter2(const int* __restrict__ src, const int* __restrict__ dst,
                           const float* __restrict__ norm,
                           const float* __restrict__ h3, float* __restrict__ out,
                           int total4) {
  int i = blockIdx.x * blockDim.x + threadIdx.x;
  if (i < total4) {
    int e = i >> 2, c0 = (i & 3) * 4;
    float nm = norm[e];
    float4 v = *(const float4*)(h3 + (size_t)src[e] * OUTD + c0);
    float* op = out + (size_t)dst[e] * OUTD + c0;
    atomicAdd(op + 0, nm * v.x);
    atomicAdd(op + 1, nm * v.y);
    atomicAdd(op + 2, nm * v.z);
    atomicAdd(op + 3, nm * v.w);
  }
}

extern "C" void kernel_launch(void* const* d_in, const int* in_sizes, int n_in,
                              void* d_out, int out_size, void* d_ws, size_t ws_size,
                              hipStream_t stream) {
  (void)n_in; (void)out_size; (void)ws_size;
  const float* x  = (const float*)d_in[0];
  const int*   ei = (const int*)d_in[1];   // [2, E] (JAX x64 off -> int32)
  const float* ew = (const float*)d_in[2];
  const float* W1 = (const float*)d_in[3];
  const float* b1 = (const float*)d_in[4];
  const float* W2 = (const float*)d_in[5];
  const float* b2 = (const float*)d_in[6];
  float* out = (float*)d_out;

  const int N = in_sizes[0] / IN_DIM;   // 100000
  const int E = in_sizes[2];            // 1600000
  const int* srcp = ei;
  const int* dstp = ei + E;

  // workspace carve-out (256B aligned)
  char* ws = (char*)d_ws;
  size_t off = 0;
  auto carve = [&](size_t bytes) -> void* {
    void* p = ws + off;
    off = (off + bytes + 255) & ~(size_t)255;
    return p;
  };
  float* dinv = (float*)carve((size_t)N * 4);
  float* norm = (float*)carve((size_t)E * 4);
  float* h1   = (float*)carve((size_t)N * HID * 4);
  float* out1 = (float*)carve((size_t)N * HID * 4);
  float* h3   = h1;  // h1 dead after scatter1 -> reuse for layer-2 output

  const int nt = (N + 15) / 16;  // 16-row tiles

  k_init_deg<<<dim3((N + 255) / 256), dim3(256), 0, stream>>>(dinv, N);
  k_deg_accum<<<dim3((E + 255) / 256), dim3(256), 0, stream>>>(dstp, ew, dinv, E);
  k_dinv<<<dim3((N + 255) / 256), dim3(256), 0, stream>>>(dinv, N);
  k_norm<<<dim3((E + 255) / 256), dim3(256), 0, stream>>>(srcp, dstp, ew, dinv, norm, E);

  k_gemm1<<<dim3(nt), dim3(256), 0, stream>>>(x, W1, h1, N);

  int t1 = N * (HID / 4);
  k_init_out1<<<dim3((t1 + 255) / 256), dim3(256), 0, stream>>>(
      (const float4*)h1, dinv, (const float4*)b1, (float4*)out1, t1);
  k_scatter1<<<dim3((E + 7) / 8), dim3(256), 0, stream>>>(srcp, dstp, norm, h1, out1, E);

  k_gemm2<<<dim3((nt + 7) / 8), dim3(256), 0, stream>>>(out1, W2, h3, N);

  int t2 = N * (OUTD / 4);
  k_init_out2<<<dim3((t2 + 255) / 256), dim3(256), 0, stream>>>(
      (const float4*)h3, dinv, (const float4*)b2, (float4*)out, t2);
  int t3 = E * 4;
  k_scatter2<<<dim3((t3 + 255) / 256), dim3(256), 0, stream>>>(srcp, dstp, norm, h3,
                                                               out, t3);
}